// GCNmodel_48928267436271
// MI455X (gfx1250) — compile-verified
//
#include <hip/hip_runtime.h>

// GCN 2-layer forward for MI455X (gfx1250, wave32).
// Bottleneck is edge gather/scatter (~1GB of L2-resident traffic);
// GEMMs use V_WMMA_F32_16X16X4_F32 to keep exact fp32 semantics.

#define N_NODES 50000
#define N_EDGES 800000
#define F_IN    64
#define F_HID   128
#define F_OUT   64

typedef __attribute__((ext_vector_type(2))) float v2f;
typedef __attribute__((ext_vector_type(8))) float v8f;

// ---------------- elementwise / setup kernels ----------------

__global__ void zero_kernel(float* __restrict__ p, int n) {
    int i = blockIdx.x * blockDim.x + threadIdx.x;
    if (i < n) p[i] = 0.0f;
}

__global__ void degree_kernel(const int* __restrict__ src, const int* __restrict__ dst,
                              float* __restrict__ out_deg, float* __restrict__ in_deg) {
    int e = blockIdx.x * blockDim.x + threadIdx.x;
    if (e < N_EDGES) {
        atomicAdd(&out_deg[src[e]], 1.0f);
        atomicAdd(&in_deg[dst[e]], 1.0f);
    }
}

__global__ void norm_kernel(const float* __restrict__ out_deg, const float* __restrict__ in_deg,
                            float* __restrict__ norm_src, float* __restrict__ norm_dst) {
    int i = blockIdx.x * blockDim.x + threadIdx.x;
    if (i < N_NODES) {
        norm_src[i] = rsqrtf(fmaxf(out_deg[i], 1.0f));
        norm_dst[i] = rsqrtf(fmaxf(in_deg[i], 1.0f));
    }
}

// agg[dst][f] += feat[src][f] * scale[src]   (64-wide features, 64 lanes/edge)
__global__ void scatter_scaled_kernel(const float* __restrict__ feat,
                                      const float* __restrict__ scale,
                                      const int* __restrict__ src, const int* __restrict__ dst,
                                      float* __restrict__ agg) {
    long long t = (long long)blockIdx.x * 256 + threadIdx.x;
    int e = (int)(t >> 6);
    int f = (int)(t & 63);
    if (e < N_EDGES) {
        int s = src[e], d = dst[e];
        atomicAdd(&agg[(long long)d * 64 + f], feat[(long long)s * 64 + f] * scale[s]);
    }
}

// agg[dst][f] += feat[src][f]   (features already scaled)
__global__ void scatter_plain_kernel(const float* __restrict__ feat,
                                     const int* __restrict__ src, const int* __restrict__ dst,
                                     float* __restrict__ agg) {
    long long t = (long long)blockIdx.x * 256 + threadIdx.x;
    int e = (int)(t >> 6);
    int f = (int)(t & 63);
    if (e < N_EDGES) {
        int s = src[e], d = dst[e];
        atomicAdd(&agg[(long long)d * 64 + f], feat[(long long)s * 64 + f]);
    }
}

// out[i][f] = agg[i][f] * norm_dst[i] + bias[f]
__global__ void final_kernel(const float* __restrict__ agg, const float* __restrict__ norm_dst,
                             const float* __restrict__ bias, float* __restrict__ out) {
    long long t = (long long)blockIdx.x * 256 + threadIdx.x;
    if (t < (long long)N_NODES * 64) {
        int i = (int)(t >> 6);
        int f = (int)(t & 63);
        out[t] = agg[t] * norm_dst[i] + bias[f];
    }
}

// ---------------- fp32 WMMA GEMM ----------------
// D[N_NODES, NDIM] = A[N_NODES, KDIM] @ B[KDIM, NDIM], one 16-row block per
// workgroup, one wave per 16-col tile. N_NODES = 3125*16 exactly -> no bounds
// divergence, EXEC is all-ones for every v_wmma as the ISA requires.
//
// Lane layouts per CDNA5 ISA 7.12.2 (wave32):
//   A 16x4  f32: lanes 0-15 hold M=lane   {K, K+1}; lanes 16-31 hold M=lane-16 {K+2, K+3}
//   B 4x16  f32: lanes 0-15 hold N=lane   {K, K+1}; lanes 16-31 hold N=lane-16 {K+2, K+3}
//   C/D 16x16 f32: VGPR r -> M = r (lanes 0-15) or r+8 (lanes 16-31), N = lane%16
template <int KDIM, int NDIM, bool SCALE_A, bool EPILOGUE>
__global__ void wmma_gemm_kernel(const float* __restrict__ A,
                                 const float* __restrict__ B,
                                 const float* __restrict__ rowscale_a,  // used if SCALE_A
                                 const float* __restrict__ rowscale_d,  // used if EPILOGUE
                                 const float* __restrict__ bias,        // used if EPILOGUE
                                 float* __restrict__ D) {
    const int lane    = threadIdx.x & 31;
    const int wave    = threadIdx.x >> 5;   // column tile index
    const int half    = lane >> 4;          // 0: K/M low half, 1: high half
    const int l       = lane & 15;
    const int row0    = blockIdx.x * 16;
    const int arow    = row0 + l;
    const int col     = wave * 16 + l;

    float ascale = 1.0f;
    if (SCALE_A) ascale = rowscale_a[arow];

    v8f c = {};
    const float* arowp = A + (long long)arow * KDIM;

    #pragma unroll
    for (int k = 0; k < KDIM; k += 4) {
        const int ka = k + half * 2;
        v2f a = *(const v2f*)(arowp + ka);          // 8B load, contiguous in K
        if (SCALE_A) { a.x *= ascale; a.y *= ascale; }
        v2f b;
        b.x = B[(long long)ka * NDIM + col];        // coalesced across lanes
        b.y = B[(long long)(ka + 1) * NDIM + col];
        c = __builtin_amdgcn_wmma_f32_16x16x4_f32(
                /*neg_a=*/false, a, /*neg_b=*/false, b,
                /*c_mod=*/(short)0, c, /*reuse_a=*/false, /*reuse_b=*/false);
    }

    #pragma unroll
    for (int r = 0; r < 8; ++r) {
        int row = row0 + r + half * 8;
        float v = c[r];
        if (EPILOGUE) {
            v = v * rowscale_d[row] + bias[col];
            v = fmaxf(v, 0.0f);                      // ReLU
        }
        D[(long long)row * NDIM + col] = v;
    }
}

// ---------------- launcher ----------------

extern "C" void kernel_launch(void* const* d_in, const int* in_sizes, int n_in,
                              void* d_out, int out_size, void* d_ws, size_t ws_size,
                              hipStream_t stream) {
    const float* x   = (const float*)d_in[0];   // [N, 64]
    const float* W1  = (const float*)d_in[1];   // [64, 128]
    const float* b1  = (const float*)d_in[2];   // [128]
    const float* W2  = (const float*)d_in[3];   // [128, 64]
    const float* b2  = (const float*)d_in[4];   // [64]
    const int*   src = (const int*)d_in[5];     // [E] (JAX default x64 off -> int32)
    const int*   dst = (const int*)d_in[6];     // [E]
    float*       out = (float*)d_out;           // [N, 64]

    // Workspace layout (floats)
    float* ws       = (float*)d_ws;
    float* out_deg  = ws;                                  // N
    float* in_deg   = out_deg + N_NODES;                   // N
    float* norm_src = in_deg + N_NODES;                    // N
    float* norm_dst = norm_src + N_NODES;                  // N
    float* agg      = norm_dst + N_NODES;                  // N*64 (reused both layers)
    float* h1       = agg + (long long)N_NODES * 64;       // N*128
    float* h2w      = h1 + (long long)N_NODES * 128;       // N*64

    const int ZT = 256;
    const long long aggElems = (long long)N_NODES * 64;
    const long long edgeThreads = (long long)N_EDGES * 64;

    // 1. degrees -> norms
    zero_kernel<<<(2 * N_NODES + ZT - 1) / ZT, ZT, 0, stream>>>(out_deg, 2 * N_NODES);
    degree_kernel<<<(N_EDGES + ZT - 1) / ZT, ZT, 0, stream>>>(src, dst, out_deg, in_deg);
    norm_kernel<<<(N_NODES + ZT - 1) / ZT, ZT, 0, stream>>>(out_deg, in_deg, norm_src, norm_dst);

    // 2. layer 1: aggregate (x * norm_src) at width 64, then GEMM + fused epilogue
    zero_kernel<<<(int)((aggElems + ZT - 1) / ZT), ZT, 0, stream>>>(agg, (int)aggElems);
    scatter_scaled_kernel<<<(int)((edgeThreads + ZT - 1) / ZT), ZT, 0, stream>>>(
        x, norm_src, src, dst, agg);
    // h1 = relu((agg @ W1) * norm_dst + b1)   [N,128], 8 col tiles -> 8 waves
    wmma_gemm_kernel<F_IN, F_HID, false, true><<<N_NODES / 16, 8 * 32, 0, stream>>>(
        agg, W1, nullptr, norm_dst, b1, h1);

    // 3. layer 2: GEMM first (A scaled by norm_src), aggregate at width 64
    zero_kernel<<<(int)((aggElems + ZT - 1) / ZT), ZT, 0, stream>>>(agg, (int)aggElems);
    // h2w = (h1 * norm_src) @ W2   [N,64], 4 col tiles -> 4 waves
    wmma_gemm_kernel<F_HID, F_OUT, true, false><<<N_NODES / 16, 4 * 32, 0, stream>>>(
        h1, W2, norm_src, nullptr, nullptr, h2w);
    scatter_plain_kernel<<<(int)((edgeThreads + ZT - 1) / ZT), ZT, 0, stream>>>(
        h2w, src, dst, agg);

    // 4. out = agg * norm_dst + b2
    final_kernel<<<(int)((aggElems + ZT - 1) / ZT), ZT, 0, stream>>>(agg, norm_dst, b2, out);
}